// BaseSmear_2388001816812
// MI455X (gfx1250) — compile-verified
//
#include <hip/hip_runtime.h>

typedef __attribute__((ext_vector_type(2))) float v2f;
typedef __attribute__((ext_vector_type(8))) float v8f;

#define II    8
#define CCH   32
#define HH    240
#define WW    320
#define HWSZ  (HH * WW)
#define CHW   (CCH * HWSZ)
#define NPTS  (64 * 64 * 64)
#define OUTC  37

// Each wave: 16 points x 8 cameras via two V_WMMA_F32_16X16X4_F32.
// A (16x4): rows M = 4*img_local + {proj0, proj1, proj2, tcw_row2}
// B (4x16): homogeneous points (x,y,z,1), one point per column.
// D (16x16): lane p holds rows 0..7 (imgs 0,1 of this wmma) for point p,
//            lane p+16 holds rows 8..15 (imgs 2,3) for point p.
__global__ __launch_bounds__(256) void proj_gather_wmma(
    const float* __restrict__ images,   // (8,32,240,320)
    const float* __restrict__ trans,    // (8,3,4)
    const float* __restrict__ tcw,      // (8,4,4)
    const float* __restrict__ coords,   // (3,N)
    float* __restrict__ out)            // (8,37,N)
{
  const int lane = threadIdx.x & 31;
  const int wave = blockIdx.x * 8 + (threadIdx.x >> 5);
  const int p    = lane & 15;          // point within 16-chunk / row M within A
  const int half = lane >> 4;          // selects K-pair for A, K rows 2,3 for B
  const int n    = wave * 16 + p;      // this lane's point index

  // ---- A operands: 16 transformation rows per WMMA (4 rows x 4 images) ----
  v2f a0, a1;
  {
    const int kb   = half * 2;         // lane<16 -> K0,K1 ; lane>=16 -> K2,K3
    const int imgl = p >> 2;           // image-within-wmma 0..3
    const int rs   = p & 3;            // 0..2 = proj rows, 3 = depth row
    const float* r0 = (rs < 3) ? (trans + imgl * 12 + rs * 4)
                               : (tcw + imgl * 16 + 2 * 4);
    a0.x = r0[kb]; a0.y = r0[kb + 1];
    const int imgh = imgl + 4;
    const float* r1 = (rs < 3) ? (trans + imgh * 12 + rs * 4)
                               : (tcw + imgh * 16 + 2 * 4);
    a1.x = r1[kb]; a1.y = r1[kb + 1];
  }

  // full xyz of this lane's point (also feeds the B operand below)
  const float px = coords[n];
  const float py = coords[NPTS + n];
  const float pz = coords[2 * NPTS + n];

  // ---- B operand: branchless select, no exec-mask divergence ----
  v2f b;
  b.x = half ? pz : px;                // K0 = x (lanes 0-15), K2 = z (lanes 16-31)
  b.y = half ? 1.0f : py;              // K1 = y,               K3 = 1
  v8f c = {};
  v8f D0 = __builtin_amdgcn_wmma_f32_16x16x4_f32(false, a0, false, b,
                                                 (short)0, c, false, false);
  v8f D1 = __builtin_amdgcn_wmma_f32_16x16x4_f32(false, a1, false, b,
                                                 (short)0, c, false, false);

  // ---- per lane: 4 (image, point) records -> gather + store ----
#pragma unroll
  for (int j = 0; j < 4; ++j) {
    const int img = 2 * half + j + ((j >= 2) ? 2 : 0); // {0,1,4,5} / {2,3,6,7}
    const float pu  = (j < 2) ? D0[4 * j + 0] : D1[4 * (j - 2) + 0];
    const float pv  = (j < 2) ? D0[4 * j + 1] : D1[4 * (j - 2) + 1];
    const float pw  = (j < 2) ? D0[4 * j + 2] : D1[4 * (j - 2) + 2];
    const float dep = (j < 2) ? D0[4 * j + 3] : D1[4 * (j - 2) + 3];

    const float wz = (__builtin_fabsf(pw) < 1e-8f) ? 1e-8f : pw;
    const float u  = pu / wz;
    const float v  = pv / wz;
    const bool valid = (u >= 0.0f) & (u <= (float)(WW - 1)) &
                       (v >= 0.0f) & (v <= (float)(HH - 1)) & (dep > 0.0f);
    const float vf = valid ? 1.0f : 0.0f;

    const int ui = (int)fminf(fmaxf(rintf(u), 0.0f), (float)(WW - 1));
    const int vi = (int)fminf(fmaxf(rintf(v), 0.0f), (float)(HH - 1));

    const float* ib = images + (size_t)img * CHW + (size_t)vi * WW + ui;
    float* ob = out + (size_t)img * OUTC * NPTS + n;

#pragma unroll
    for (int ch = 0; ch < CCH; ++ch) {
      __builtin_nontemporal_store(ib[(size_t)ch * HWSZ] * vf, ob + (size_t)ch * NPTS);
    }
    __builtin_nontemporal_store(dep, ob + (size_t)32 * NPTS);
    __builtin_nontemporal_store(vf,  ob + (size_t)33 * NPTS);

    // camera center: c_a = -sum_b R[b][a] * t[b]
    const float* T = tcw + img * 16;
    const float cx = -(T[0] * T[3] + T[4] * T[7] + T[8]  * T[11]);
    const float cy = -(T[1] * T[3] + T[5] * T[7] + T[9]  * T[11]);
    const float cz = -(T[2] * T[3] + T[6] * T[7] + T[10] * T[11]);
    const float dx = px - cx, dy = py - cy, dz = pz - cz;
    const float inv = 1.0f / (sqrtf(dx * dx + dy * dy + dz * dz) + 1e-8f);
    __builtin_nontemporal_store(dx * inv, ob + (size_t)34 * NPTS);
    __builtin_nontemporal_store(dy * inv, ob + (size_t)35 * NPTS);
    __builtin_nontemporal_store(dz * inv, ob + (size_t)36 * NPTS);
  }
}

extern "C" void kernel_launch(void* const* d_in, const int* in_sizes, int n_in,
                              void* d_out, int out_size, void* d_ws, size_t ws_size,
                              hipStream_t stream) {
  const float* images = (const float*)d_in[0];
  const float* trans  = (const float*)d_in[1];
  const float* tcw    = (const float*)d_in[2];
  const float* coords = (const float*)d_in[3];
  float* out = (float*)d_out;

  // grid output: 8 * 37 * 64^3 floats, then coordinates passthrough (3 * 64^3)
  const size_t grid_elems = (size_t)II * OUTC * NPTS;

  // 16 points per wave, 8 waves per block -> 128 points per block
  const int blocks = NPTS / 128;
  proj_gather_wmma<<<blocks, 256, 0, stream>>>(images, trans, tcw, coords, out);

  hipMemcpyAsync(out + grid_elems, coords, (size_t)3 * NPTS * sizeof(float),
                 hipMemcpyDeviceToDevice, stream);
}